// KeypointDetector_62895501082654
// MI455X (gfx1250) — compile-verified
//
#include <hip/hip_runtime.h>
#include <hip/hip_bf16.h>
#include <stdint.h>

// Problem constants (from reference)
#define NB     16          // batches
#define NPTS   16384       // points per cloud
#define NCH    128         // feature channels
#define NSAMP  1024        // samples
#define NSEM   20          // semantic classes
#define NTHR   1024        // threads per block (32 wave32 waves, 1 block per WGP)
#define PPT    (NPTS/NTHR) // 16 points per thread, contiguous chunk per thread

// LDS layout (bytes) — total ~201 KB, fits 320 KB WGP LDS
#define LDS_XYZ_OFF   0
#define LDS_XYZ_BYTES (NPTS*3*4)            // 196608: staged xyz slab (TDM dest)
#define LDS_KEY_OFF   (LDS_XYZ_OFF + LDS_XYZ_BYTES)   // 196608 (8B aligned)
#define LDS_CNT_OFF   (LDS_KEY_OFF + 32)    // 3x u64 rotating argmax keys (+pad)
#define LDS_SEL_OFF   (LDS_CNT_OFF + 128)   // 20-bin histogram (+pad)
#define LDS_TOTAL     (LDS_SEL_OFF + NSAMP*4) // + selected-index list = 200864 B

extern "C" __global__ __launch_bounds__(NTHR)
void wfps_fused_kernel(const float* __restrict__ xyz,
                       const float* __restrict__ feat,
                       const int*   __restrict__ lab,
                       float* __restrict__ out)
{
    extern __shared__ char smem[];
    float*              s_xyz = (float*)(smem + LDS_XYZ_OFF);
    unsigned long long* s_key = (unsigned long long*)(smem + LDS_KEY_OFF);
    int*                s_cnt = (int*)(smem + LDS_CNT_OFF);
    int*                s_sel = (int*)(smem + LDS_SEL_OFF);

    const int b    = blockIdx.x;
    const int tid  = threadIdx.x;
    const int lane = tid & 31;
    const int base = tid * PPT;                 // contiguous chunk -> tie-break == lowest index

    const float* gxyz = xyz + (size_t)b * NPTS * 3;
    const int*   glab = lab + (size_t)b * NPTS;

    // ---------------- stage xyz -> LDS -----------------------------------
#if defined(__gfx1250__) && __has_builtin(__builtin_amdgcn_tensor_load_to_lds)
    // CDNA5 Tensor Data Mover: one wave issues a DMA of the 192KB xyz slab
    // into LDS; it runs concurrently with the histogram below.
    if (tid < 32) { // wave 0 only (TDM is a wave-level op, EXEC ignored)
        typedef unsigned int u32x4 __attribute__((ext_vector_type(4)));
        typedef int          i32x4 __attribute__((ext_vector_type(4)));
        typedef int          i32x8 __attribute__((ext_vector_type(8)));
        unsigned long long ga  = (unsigned long long)(const void*)gxyz;
        unsigned           lda = (unsigned)(unsigned long long)(void*)s_xyz; // LDS byte offset = low 32b of generic addr
        // D# group0: count=1 | lds_addr | global_addr[56:0] | type=2
        u32x4 g0 = { 1u,
                     lda,
                     (unsigned)(ga & 0xFFFFFFFFull),
                     ((unsigned)((ga >> 32) & 0x01FFFFFFull)) | 0x80000000u };
        // D# group1: 2D tensor 12288 x 4 floats (data_size=4B), tile = whole tensor
        i32x8 g1 = { (int)0x00020000u,            // wg_mask=0, data_size=2 (4B)
                     (int)(12288u << 16),          // tensor_dim0[15:0] in dw1[31:16]
                     (int)(4u << 16),              // tensor_dim0[31:16]=0 | tensor_dim1[15:0]=4
                     (int)(12288u << 16),          // tensor_dim1[31:16]=0 | tile_dim0=12288
                     (int)4,                       // tile_dim1=4, tile_dim2=0
                     (int)12288,                   // tensor_dim0_stride[31:0]
                     0, 0 };
        i32x4 gz4 = { 0, 0, 0, 0 };                // groups 2/3 unused (<=2D)
        i32x8 gz8 = { 0, 0, 0, 0, 0, 0, 0, 0 };    // trailing group (6-arg form)
        __builtin_amdgcn_tensor_load_to_lds(g0, g1, gz4, gz4, gz8, 0);
    }
#else
    for (int j = tid; j < NPTS * 3; j += NTHR) s_xyz[j] = gxyz[j];
#endif

    // ---------------- histogram -> per-point weights (overlaps TDM) ------
    int l[PPT];
#pragma unroll
    for (int i = 0; i < PPT; ++i) l[i] = glab[base + i];

    if (tid < NSEM) s_cnt[tid] = 0;
    if (tid == 0) { s_key[0] = 0ull; s_key[1] = 0ull; s_key[2] = 0ull; }
    __syncthreads();

#pragma unroll
    for (int i = 0; i < PPT; ++i) atomicAdd(&s_cnt[l[i]], 1);
    __syncthreads();

    float w[PPT];
#pragma unroll
    for (int i = 0; i < PPT; ++i) w[i] = (float)s_cnt[l[i]];

#if defined(__gfx1250__) && __has_builtin(__builtin_amdgcn_tensor_load_to_lds)
    if (tid < 32) __builtin_amdgcn_s_wait_tensorcnt(0);  // wave0: TDM done
#endif
    __syncthreads();  // xyz slab now visible to all waves

    // ---------------- register-resident FPS state ------------------------
    float x[PPT], y[PPT], z[PPT], md[PPT];
#pragma unroll
    for (int i = 0; i < PPT; ++i) {
        x[i]  = s_xyz[(base + i) * 3 + 0];
        y[i]  = s_xyz[(base + i) * 3 + 1];
        z[i]  = s_xyz[(base + i) * 3 + 2];
        md[i] = 1e10f;
    }
    float px = s_xyz[0], py = s_xyz[1], pz = s_xyz[2];   // first pick = index 0
    if (tid == 0) s_sel[0] = 0;
    __syncthreads();  // s_key slots zeroed + everyone has px before first round

    // ---------------- 1023 sequential weighted-FPS rounds ----------------
    for (int it = 1; it < NSAMP; ++it) {
        const int slot = it % 3;

        // local: distance to last pick, running min, weighted argmax (first-max kept)
        float bv = -1.0f; int bi = base;
#pragma unroll
        for (int i = 0; i < PPT; ++i) {
            float dx = x[i] - px, dy = y[i] - py, dz = z[i] - pz;
            float d  = dx * dx + dy * dy + dz * dz;
            float m  = fminf(md[i], d);
            md[i] = m;
            float sc = m * w[i];
            if (sc > bv) { bv = sc; bi = base + i; }
        }

        // pack (score, index): u64 order == (max score, then lowest index)
        unsigned long long key =
            ((unsigned long long)__float_as_uint(bv) << 32) |
            (unsigned long long)(0xFFFFFFFFu - (unsigned)bi);

        // wave32 xor-butterfly reduction
#pragma unroll
        for (int off = 16; off > 0; off >>= 1) {
            unsigned long long o = __shfl_xor(key, off, 32);
            key = (o > key) ? o : key;
        }
        if (lane == 0) atomicMax(&s_key[slot], key);   // ds_max_u64, 32 waves
        __syncthreads();                                // single barrier per round

        unsigned long long kk = s_key[slot];
        int idx = (int)(0xFFFFFFFFu - (unsigned)(kk & 0xFFFFFFFFull));
        px = s_xyz[idx * 3 + 0];                        // LDS broadcast reads
        py = s_xyz[idx * 3 + 1];
        pz = s_xyz[idx * 3 + 2];
        if (tid == 0) {
            s_sel[it] = idx;
            s_key[(it + 2) % 3] = 0ull;  // reset slot idle this window (race-free)
        }
    }
    __syncthreads();

    // ---------------- fused gathers ---------------------------------------
    float* out0 = out;                                        // [B,1024,3]
    float* out1 = out + (size_t)NB * NSAMP * 3;               // [B,128,1024]
    int*   out2 = (int*)(out + (size_t)NB * NSAMP * 3 + (size_t)NB * NCH * NSAMP); // [B,1024] i32

    const int sidx = s_sel[tid];   // one sample per thread

    // sampled xyz (from LDS slab)
    float* o0 = out0 + (size_t)b * NSAMP * 3 + (size_t)tid * 3;
    o0[0] = s_xyz[sidx * 3 + 0];
    o0[1] = s_xyz[sidx * 3 + 1];
    o0[2] = s_xyz[sidx * 3 + 2];

    // sampled labels
    out2[(size_t)b * NSAMP + tid] = glab[sidx];

    // sampled features: scattered reads, fully coalesced writes
    const float* fb = feat + (size_t)b * NCH * NPTS;
    float*       ob = out1 + (size_t)b * NCH * NSAMP;
#pragma unroll 4
    for (int c = 0; c < NCH; ++c)
        ob[(size_t)c * NSAMP + tid] = fb[(size_t)c * NPTS + sidx];
}

extern "C" void kernel_launch(void* const* d_in, const int* in_sizes, int n_in,
                              void* d_out, int out_size, void* d_ws, size_t ws_size,
                              hipStream_t stream) {
    const float* xyz  = (const float*)d_in[0];   // [16,16384,3]  f32
    const float* feat = (const float*)d_in[1];   // [16,128,16384] f32
    const int*   lab  = (const int*)d_in[2];     // [16,16384]    i32
    float* out = (float*)d_out;

    // ~201 KB dynamic LDS per workgroup (WGP holds 320 KB); opt in above 64 KB.
    (void)hipFuncSetAttribute(reinterpret_cast<const void*>(wfps_fused_kernel),
                              hipFuncAttributeMaxDynamicSharedMemorySize, LDS_TOTAL);

    wfps_fused_kernel<<<dim3(NB), dim3(NTHR), LDS_TOTAL, stream>>>(xyz, feat, lab, out);
}